// HyperMSG_58042188038707
// MI455X (gfx1250) — compile-verified
//
#include <hip/hip_runtime.h>
#include <math.h>

// ---------------------------------------------------------------------------
// HyperMSG on MI455X (gfx1250, wave32).
//
// Algebraic restructuring: all aggregation steps in hconv are row-linear, so
//   agg(H) @ W1 == agg(H @ W1).
// We project H @ W1 (the only large GEMM -> V_WMMA_F32_16X16X4_F32) FIRST,
// then do all scatter/gather at d=16 instead of d=256, cutting atomic and
// gather traffic by 16x. Layer-2 projection (16x40) + log_softmax is fused
// into the final per-node kernel.
// ---------------------------------------------------------------------------

#define N_NODES 100000
#define N_EDGES 20000
#define N_INC   800000
#define D_IN    256
#define D_HID   16
#define D_OUT   40

typedef __attribute__((ext_vector_type(2))) float v2f;
typedef __attribute__((ext_vector_type(8))) float v8f;

// ---------------- zero fill ----------------
__global__ void zero_f32(float* __restrict__ p, long n) {
  long i = (long)blockIdx.x * blockDim.x + threadIdx.x;
  if (i < n) p[i] = 0.0f;
}

// ---------------- H1 = H @ W1, fp32 WMMA 16x16x4 ----------------
// One wave per 16-row M tile. W1 (256x16 = 16KB) staged in LDS.
// A layout (16x4 f32): lane = M + 16*(K>>1), reg = K&1.
// B layout (4x16 f32): lane = N + 16*(K>>1), reg = K&1.
// C/D (16x16 f32):     lane = N + 16*(M>=8), reg = M%8.
__global__ void __launch_bounds__(256)
gemm_h_w1(const float* __restrict__ H, const float* __restrict__ W1,
          float* __restrict__ H1) {
  __shared__ float sW[D_IN * D_HID];  // 16 KB
  for (int i = threadIdx.x; i < D_IN * D_HID; i += 256) sW[i] = W1[i];
  __syncthreads();

  const int lane = threadIdx.x & 31;
  const int wave = threadIdx.x >> 5;
  const int tile = blockIdx.x * 8 + wave;           // M-tile index
  if (tile >= (N_NODES / 16)) return;               // whole-wave uniform exit

  const int m0 = tile * 16;
  const int mn = lane & 15;                          // A: row-in-tile; B: col
  const int kh = (lane >> 4) << 1;                   // 0 for lanes 0-15, 2 else
  const float* __restrict__ Arow = H + (long)(m0 + mn) * D_IN;

  v8f c = {0.f, 0.f, 0.f, 0.f, 0.f, 0.f, 0.f, 0.f};
#pragma unroll 4
  for (int k0 = 0; k0 < D_IN; k0 += 4) {
    const int kb = k0 + kh;
    v2f a, b;
    a.x = Arow[kb];
    a.y = Arow[kb + 1];
    b.x = sW[kb * D_HID + mn];
    b.y = sW[(kb + 1) * D_HID + mn];
    c = __builtin_amdgcn_wmma_f32_16x16x4_f32(false, a, false, b,
                                              (short)0, c, false, false);
  }

  const int rbase = m0 + ((lane >> 4) << 3);
#pragma unroll
  for (int r = 0; r < 8; ++r)
    H1[(long)(rbase + r) * D_HID + mn] = c[r];
}

// ---------------- degree accumulation (layer-agnostic, done once) ----------
__global__ void degrees_kernel(const int* __restrict__ node_idx,
                               const int* __restrict__ edge_idx,
                               const float* __restrict__ w,
                               float* __restrict__ e_den,
                               float* __restrict__ v_deg) {
  int i = blockIdx.x * blockDim.x + threadIdx.x;
  if (i >= N_INC) return;
  const int n = node_idx[i];
  const int e = edge_idx[i];
  atomicAdd(&e_den[e], w[n]);
  atomicAdd(&v_deg[n], 1.0f);
}

// ---------------- node -> edge scatter, 16 features, w-weighted ------------
__global__ void scatter_edge16(const int* __restrict__ node_idx,
                               const int* __restrict__ edge_idx,
                               const float* __restrict__ w,
                               const float* __restrict__ X16,
                               float* __restrict__ e_sum) {
  long t = (long)blockIdx.x * blockDim.x + threadIdx.x;
  long i = t >> 4;
  int  f = (int)(t & 15);
  if (i >= N_INC) return;
  const int n = node_idx[i];
  const int e = edge_idx[i];
  atomicAdd(&e_sum[(long)e * D_HID + f], X16[(long)n * D_HID + f] * w[n]);
}

// ---------------- e_feat = e_sum / max(e_den, 1e-6), in place ---------------
__global__ void edge_div16(float* __restrict__ e_sum,
                           const float* __restrict__ e_den) {
  long t = (long)blockIdx.x * blockDim.x + threadIdx.x;
  if (t >= (long)N_EDGES * D_HID) return;
  float den = e_den[t >> 4];
  den = den > 1e-6f ? den : 1e-6f;
  e_sum[t] = e_sum[t] / den;
}

// ---------------- edge -> node scatter, 16 features -------------------------
__global__ void scatter_node16(const int* __restrict__ node_idx,
                               const int* __restrict__ edge_idx,
                               const float* __restrict__ e_feat,
                               float* __restrict__ v_sum) {
  long t = (long)blockIdx.x * blockDim.x + threadIdx.x;
  long i = t >> 4;
  int  f = (int)(t & 15);
  if (i >= N_INC) return;
  const int n = node_idx[i];
  const int e = edge_idx[i];
  atomicAdd(&v_sum[(long)n * D_HID + f], e_feat[(long)e * D_HID + f]);
}

// ---------------- layer-1 node stage: x = relu(v_sum/deg + b1) --------------
__global__ void node_relu16(const float* __restrict__ v_sum,
                            const float* __restrict__ v_deg,
                            const float* __restrict__ b1,
                            float* __restrict__ x) {
  long t = (long)blockIdx.x * blockDim.x + threadIdx.x;
  if (t >= (long)N_NODES * D_HID) return;
  float deg = v_deg[t >> 4];
  deg = deg > 1.0f ? deg : 1.0f;
  float v = v_sum[t] / deg + b1[t & 15];
  x[t] = v > 0.0f ? v : 0.0f;
}

// ---------------- final: (v_sum2/deg) @ W2 + b2, log_softmax ----------------
__global__ void __launch_bounds__(256)
final_proj_lsm(const float* __restrict__ v_sum2,
               const float* __restrict__ v_deg,
               const float* __restrict__ W2,
               const float* __restrict__ b2,
               float* __restrict__ out) {
  __shared__ float sW[D_HID * D_OUT];  // 2.5 KB
  __shared__ float sb[D_OUT];
  for (int i = threadIdx.x; i < D_HID * D_OUT; i += 256) sW[i] = W2[i];
  if (threadIdx.x < D_OUT) sb[threadIdx.x] = b2[threadIdx.x];
  __syncthreads();

  const int nd = blockIdx.x * blockDim.x + threadIdx.x;
  if (nd >= N_NODES) return;

  float deg = v_deg[nd];
  deg = deg > 1.0f ? deg : 1.0f;
  const float inv = 1.0f / deg;

  float v[D_HID];
#pragma unroll
  for (int f = 0; f < D_HID; ++f) v[f] = v_sum2[(long)nd * D_HID + f] * inv;

  float lg[D_OUT];
  float mx = -3.4e38f;
#pragma unroll
  for (int j = 0; j < D_OUT; ++j) {
    float acc = sb[j];
#pragma unroll
    for (int f = 0; f < D_HID; ++f) acc = fmaf(v[f], sW[f * D_OUT + j], acc);
    lg[j] = acc;
    mx = acc > mx ? acc : mx;
  }
  float s = 0.0f;
#pragma unroll
  for (int j = 0; j < D_OUT; ++j) s += expf(lg[j] - mx);
  const float lse = mx + logf(s);
#pragma unroll
  for (int j = 0; j < D_OUT; ++j) out[(long)nd * D_OUT + j] = lg[j] - lse;
}

// ---------------------------------------------------------------------------
extern "C" void kernel_launch(void* const* d_in, const int* in_sizes, int n_in,
                              void* d_out, int out_size, void* d_ws, size_t ws_size,
                              hipStream_t stream) {
  const float* H        = (const float*)d_in[0];   // [100000,256]
  const float* w        = (const float*)d_in[1];   // [100000]
  const int*   node_idx = (const int*)  d_in[2];   // [800000]
  const int*   edge_idx = (const int*)  d_in[3];   // [800000]
  const float* W1       = (const float*)d_in[4];   // [256,16]
  const float* b1       = (const float*)d_in[5];   // [16]
  const float* W2       = (const float*)d_in[6];   // [16,40]
  const float* b2       = (const float*)d_in[7];   // [40]
  float* out = (float*)d_out;                      // [100000,40]

  // Workspace layout (floats). Accumulators placed first for one zero pass.
  float* ws     = (float*)d_ws;
  float* e_sum1 = ws;                        // 320000   (becomes e_feat1)
  float* e_den  = e_sum1 + 320000;           // 20000
  float* v_deg  = e_den  + 20000;            // 100000
  float* v_sum1 = v_deg  + 100000;           // 1600000
  float* e_sum2 = v_sum1 + 1600000;          // 320000   (becomes e_feat2)
  float* v_sum2 = e_sum2 + 320000;           // 1600000
  float* H1     = v_sum2 + 1600000;          // 1600000  H @ W1
  float* xbuf   = H1     + 1600000;          // 1600000  layer-1 activations
  const long zcount = 320000 + 20000 + 100000 + 1600000 + 320000 + 1600000;

  const int B = 256;
  // 0) zero all accumulators
  zero_f32<<<(int)((zcount + B - 1) / B), B, 0, stream>>>(ws, zcount);

  // 1) H1 = H @ W1   (WMMA f32 16x16x4; 6250 M-tiles, 8 waves/block)
  gemm_h_w1<<<(N_NODES / 16 + 7) / 8, B, 0, stream>>>(H, W1, H1);

  // 2) degrees (shared by both layers)
  degrees_kernel<<<(N_INC + B - 1) / B, B, 0, stream>>>(node_idx, edge_idx, w,
                                                        e_den, v_deg);

  const long incT  = (long)N_INC * D_HID;          // 12.8M scatter threads
  const int  gInc  = (int)((incT + B - 1) / B);
  const int  gEdge = (int)(((long)N_EDGES * D_HID + B - 1) / B);
  const int  gNode = (int)(((long)N_NODES * D_HID + B - 1) / B);

  // ---- layer 1 (features already projected to d=16) ----
  scatter_edge16<<<gInc, B, 0, stream>>>(node_idx, edge_idx, w, H1, e_sum1);
  edge_div16    <<<gEdge, B, 0, stream>>>(e_sum1, e_den);
  scatter_node16<<<gInc, B, 0, stream>>>(node_idx, edge_idx, e_sum1, v_sum1);
  node_relu16   <<<gNode, B, 0, stream>>>(v_sum1, v_deg, b1, xbuf);

  // ---- layer 2 (aggregate at d=16, project to 40 in final kernel) ----
  scatter_edge16<<<gInc, B, 0, stream>>>(node_idx, edge_idx, w, xbuf, e_sum2);
  edge_div16    <<<gEdge, B, 0, stream>>>(e_sum2, e_den);
  scatter_node16<<<gInc, B, 0, stream>>>(node_idx, edge_idx, e_sum2, v_sum2);

  // ---- fused W2 projection + log_softmax ----
  final_proj_lsm<<<(N_NODES + B - 1) / B, B, 0, stream>>>(v_sum2, v_deg, W2, b2, out);
}